// NURBS2D_66383014527123
// MI455X (gfx1250) — compile-verified
//
#include <hip/hip_runtime.h>

// NURBS surface eval reformulated as banded-basis GEMMs for CDNA5 WMMA.
// Sizes fixed per reference: NU=NV=64, M=N=1024, DEG=3.

typedef __attribute__((ext_vector_type(2))) float v2f;
typedef __attribute__((ext_vector_type(8))) float v8f;

// Workspace layout (float offsets)
#define OFF_CTRLH 0        // 64*64*4        = 16384
#define OFF_A     16384    // 1024*64        = 65536
#define OFF_Q     81920    // 4*32*1024*2    = 262144  (K-pair interleaved)
#define OFF_NU    344064   // 4*1024
#define OFF_NV    348160   // 4*1024
#define OFF_USPAN 352256   // 1024 ints
#define OFF_VSPAN 353280   // 1024 ints

// Normalized clamped-uniform knot: Uk[i] = clamp(i-3, 0, 61)/61
__device__ __forceinline__ float knot_at(int idx) {
  int k = idx - 3;
  k = k < 0 ? 0 : (k > 61 ? 61 : k);
  return (float)k * (1.0f / 61.0f);
}

// Replicates reference _spans: argmin over inner knots jj/61, jj=0..61
__device__ __forceinline__ int find_span(float t) {
  float best = 3.4e38f;
  int bestj = 0;
  for (int jj = 0; jj <= 61; ++jj) {
    float d = t - (float)jj * (1.0f / 61.0f);
    float m = (d > 1e-8f) ? d : 1.0f;
    if (m < best) { best = m; bestj = jj; }
  }
  return bestj + 3;
}

// Cox-de Boor, exact translation of reference _basis (deg=3)
__device__ __forceinline__ void basis_funcs(float t, int span, float* Nb) {
  float Ul[6];
#pragma unroll
  for (int i = -2; i <= 3; ++i) Ul[i + 2] = knot_at(span + i);
  Nb[0] = 1.0f; Nb[1] = 0.0f; Nb[2] = 0.0f; Nb[3] = 0.0f;
#pragma unroll
  for (int k = 1; k <= 3; ++k) {
    float saved = 0.0f;
#pragma unroll
    for (int r = 0; r < 3; ++r) {
      if (r >= k) break;
      float U1 = Ul[r + 3];
      float U2 = Ul[r + 3 - k];
      float den = (U1 - t) + (t - U2);
      float temp = (den == 0.0f) ? 1e-4f : (Nb[r] / den);
      Nb[r] = saved + (U1 - t) * temp;
      saved = (t - U2) * temp;
    }
    Nb[k] = saved;
  }
}

// Kernel 1: homogenize control pts, bitonic-sort u, bases/spans, banded A.
__global__ __launch_bounds__(1024) void nurbs_setup(
    const float* __restrict__ cp, const float* __restrict__ w,
    const float* __restrict__ u_in, const float* __restrict__ v_in,
    float* __restrict__ ws) {
  __shared__ float s[1024];
  const int t = threadIdx.x;
  float* ctrlh = ws + OFF_CTRLH;
  float* Amat  = ws + OFF_A;
  float* Nu    = ws + OFF_NU;
  float* Nv    = ws + OFF_NV;
  int*   uspan = (int*)(ws + OFF_USPAN);
  int*   vspan = (int*)(ws + OFF_VSPAN);

  // homogeneous control points [x*w, y*w, z*w, w]
  for (int idx = t; idx < 64 * 64; idx += 1024) {
    float wt = w[idx];
    ctrlh[idx * 4 + 0] = cp[idx * 3 + 0] * wt;
    ctrlh[idx * 4 + 1] = cp[idx * 3 + 1] * wt;
    ctrlh[idx * 4 + 2] = cp[idx * 3 + 2] * wt;
    ctrlh[idx * 4 + 3] = wt;
  }

  // bitonic sort of u (1024 = 2^10), one element per thread in LDS
  s[t] = u_in[t];
  for (unsigned k = 2; k <= 1024; k <<= 1) {
    for (unsigned j = k >> 1; j > 0; j >>= 1) {
      __syncthreads();
      unsigned ixj = (unsigned)t ^ j;
      if (ixj > (unsigned)t) {
        float a = s[t], b = s[ixj];
        bool asc = ((t & k) == 0);
        if (asc ? (a > b) : (a < b)) { s[t] = b; s[ixj] = a; }
      }
    }
  }
  __syncthreads();

  // u basis + banded A row (4 non-zeros among 64)
  {
    float tu = s[t];
    int sp = find_span(tu);
    float Nb[4];
    basis_funcs(tu, sp, Nb);
    uspan[t] = sp;
#pragma unroll
    for (int l = 0; l < 4; ++l) Nu[l * 1024 + t] = Nb[l];
    float* arow = Amat + t * 64;
    for (int i = 0; i < 64; ++i) arow[i] = 0.0f;
#pragma unroll
    for (int l = 0; l < 4; ++l) {
      int i = sp - 3 + l;
      if (i >= 0 && i < 64) arow[i] = Nb[l];
    }
  }

  // v basis (v not sorted)
  {
    float tv = v_in[t];
    int sp = find_span(tv);
    float Nb[4];
    basis_funcs(tv, sp, Nb);
    vspan[t] = sp;
#pragma unroll
    for (int l = 0; l < 4; ++l) Nv[l * 1024 + t] = Nb[l];
  }
}

// Kernel 2: Q[c][i][n] = sum_r ctrlh[i][vspan[n]-3+r][c] * Nv[r][n]
// Stored K-pair interleaved for b64 B-fragment loads in the WMMA kernel:
//   Qp[((c*32 + i/2)*1024 + n)*2 + (i&1)]
__global__ __launch_bounds__(256) void nurbs_stage1(float* __restrict__ ws) {
  const float* ctrlh = ws + OFF_CTRLH;
  const float* Nv = ws + OFF_NV;
  const int* vspan = (const int*)(ws + OFF_VSPAN);
  float* Qp = ws + OFF_Q;
  int tid = blockIdx.x * 256 + threadIdx.x;  // 0..65535
  int i = tid >> 10;                         // K index 0..63
  int n = tid & 1023;
  int vs = vspan[n] - 3;
  if (vs < 0) vs = 0;
  if (vs > 60) vs = 60;
  float nv0 = Nv[n], nv1 = Nv[1024 + n], nv2 = Nv[2048 + n], nv3 = Nv[3072 + n];
  const float* base = ctrlh + (i * 64 + vs) * 4;
  const int kp = i >> 1, ko = i & 1;
#pragma unroll
  for (int c = 0; c < 4; ++c) {
    float acc = base[c] * nv0 + base[4 + c] * nv1 + base[8 + c] * nv2 + base[12 + c] * nv3;
    Qp[((c * 32 + kp) * 1024 + n) * 2 + ko] = acc;
  }
}

// Kernel 3: surf[c] = A(1024x64) x Q[c](64x1024) via v_wmma_f32_16x16x4_f32,
// then perspective divide in-register and store xyz.
// One wave per 16x16 output tile; 4 accumulators (xyz w) share the A fragment.
// All fragment loads are single global_load_b64 (A pairs contiguous in the row,
// B pairs contiguous in the interleaved Q layout).
__global__ __launch_bounds__(256) void nurbs_wmma(const float* __restrict__ ws,
                                                  float* __restrict__ out) {
  const float* Amat = ws + OFF_A;
  const float* Qp   = ws + OFF_Q;
  const int lane = threadIdx.x & 31;
  const int wave = threadIdx.x >> 5;
  const int tile = blockIdx.x * 8 + wave;   // 4096 tiles = 64x64
  const int m0 = (tile >> 6) * 16;
  const int n0 = (tile & 63) * 16;
  const int row = lane & 15;                // M for A-frag, N for B/D-frags
  const int kh  = lane >> 4;                // selects K pair {0,1} vs {2,3}

  v8f acc0 = {}, acc1 = {}, acc2 = {}, acc3 = {};
  const v2f* arow = (const v2f*)(Amat + (m0 + row) * 64) + kh;   // A[m][64], 8B aligned
  const v2f* qrow = (const v2f*)Qp + (n0 + row);                 // Qp pairs, 8B aligned

#pragma unroll 4
  for (int kk = 0; kk < 16; ++kk) {
    // A fragment 16x4 f32: lane row = M, VGPR0/1 = K pair (kh*2, kh*2+1)
    v2f a = arow[kk * 2];
    // B fragment 4x16 f32: lane row = N, VGPR0/1 = K pair -> one b64 each
    const int kp = kk * 2 + kh;
    v2f b0 = qrow[(0 * 32 + kp) * 1024];
    v2f b1 = qrow[(1 * 32 + kp) * 1024];
    v2f b2 = qrow[(2 * 32 + kp) * 1024];
    v2f b3 = qrow[(3 * 32 + kp) * 1024];
    acc0 = __builtin_amdgcn_wmma_f32_16x16x4_f32(false, a, false, b0, (short)0, acc0, false, false);
    acc1 = __builtin_amdgcn_wmma_f32_16x16x4_f32(false, a, false, b1, (short)0, acc1, false, false);
    acc2 = __builtin_amdgcn_wmma_f32_16x16x4_f32(false, a, false, b2, (short)0, acc2, false, false);
    acc3 = __builtin_amdgcn_wmma_f32_16x16x4_f32(false, a, false, b3, (short)0, acc3, false, false);
  }

  // D layout: element j -> (M = j + kh*8, N = row); divide xyz by w and store.
  const int n = n0 + row;
#pragma unroll
  for (int j = 0; j < 8; ++j) {
    int m = m0 + j + kh * 8;
    float wv = acc3[j];
    float* o = out + (m * 1024 + n) * 3;
    o[0] = acc0[j] / wv;
    o[1] = acc1[j] / wv;
    o[2] = acc2[j] / wv;
  }
}

extern "C" void kernel_launch(void* const* d_in, const int* in_sizes, int n_in,
                              void* d_out, int out_size, void* d_ws, size_t ws_size,
                              hipStream_t stream) {
  const float* cp = (const float*)d_in[0];   // (1,64,64,3)
  const float* w  = (const float*)d_in[1];   // (1,64,64,1)
  const float* u  = (const float*)d_in[2];   // (1024,)
  const float* v  = (const float*)d_in[3];   // (1024,)
  float* ws  = (float*)d_ws;
  float* out = (float*)d_out;                // (1024,1024,3) f32

  nurbs_setup <<<1,   1024, 0, stream>>>(cp, w, u, v, ws);
  nurbs_stage1<<<256, 256,  0, stream>>>(ws);
  nurbs_wmma  <<<512, 256,  0, stream>>>(ws, out);
}